// EdgePropertyPredictionModel_13967233647441
// MI455X (gfx1250) — compile-verified
//
#include <hip/hip_runtime.h>
#include <hip/hip_bf16.h>
#include <math.h>

typedef float v2f __attribute__((ext_vector_type(2)));
typedef float v8f __attribute__((ext_vector_type(8)));

#define EMBED_D 64

// ---------------------------------------------------------------------------
// Degree accumulation: one thread per edge, float atomics into L2-resident
// 200KB arrays.
// ---------------------------------------------------------------------------
__global__ void k_degree(const int* __restrict__ src, const int* __restrict__ dst,
                         float* __restrict__ deg_out, float* __restrict__ deg_in, int E) {
    int e = blockIdx.x * blockDim.x + threadIdx.x;
    if (e < E) {
        atomicAdd(&deg_out[src[e]], 1.0f);
        atomicAdd(&deg_in[dst[e]], 1.0f);
    }
}

// In-place deg -> deg^-0.5 with min-clamp 1 (DGL norm='both').
__global__ void k_inv_sqrt(float* __restrict__ a, float* __restrict__ b, int n) {
    int i = blockIdx.x * blockDim.x + threadIdx.x;
    if (i < n) {
        a[i] = rsqrtf(fmaxf(a[i], 1.0f));
        b[i] = rsqrtf(fmaxf(b[i], 1.0f));
    }
}

// ---------------------------------------------------------------------------
// Edge scatter: agg[dst] += h[src] * inv_out[src].
// One thread per (edge, dim); 64 consecutive threads share one edge so the
// feature row read is one coalesced 256B burst and src/dst loads broadcast.
// Both h and agg are L2-resident (12.8MB each vs 192MB L2).
// ---------------------------------------------------------------------------
__global__ void k_scatter(const float* __restrict__ h, const float* __restrict__ inv_out,
                          const int* __restrict__ src, const int* __restrict__ dst,
                          float* __restrict__ agg, int E) {
    int idx = blockIdx.x * blockDim.x + threadIdx.x;   // up to E*64 = 51.2M < 2^31
    int e = idx >> 6;
    int d = idx & 63;
    if (e < E) {
        int s = src[e];
        int t = dst[e];
        float v = h[(size_t)s * EMBED_D + d] * inv_out[s];
        atomicAdd(&agg[(size_t)t * EMBED_D + d], v);
    }
}

// ---------------------------------------------------------------------------
// WMMA f32 GEMM: out[M][64] = (diag(rowScale) * A[M][K]) @ W[K][64] + bias.
// One wave32 per 16-row stripe; 4 N-tiles held in 4 v8f accumulators so the
// A fragment is loaded once per k-step. Uses V_WMMA_F32_16X16X4_F32
// (full f32 precision; 16 k-steps for K=64, 8 for K=32).
//
// A 16x4 f32 frag (ISA 7.12.2): lanes 0-15 hold M=lane, {K=k0,k0+1} in
// {v0,v1}; lanes 16-31 hold {K=k0+2,k0+3}. B 4x16 mirrored (row striped
// across lanes). C/D: 8 VGPRs, lanes 16-31 are rows M+8.
// ---------------------------------------------------------------------------
__global__ void k_gemm_wmma(const float* __restrict__ A, const float* __restrict__ rowScale,
                            const float* __restrict__ W, const float* __restrict__ bias,
                            float* __restrict__ out, int M, int K) {
    const int lane    = threadIdx.x & 31;
    const int wave    = threadIdx.x >> 5;
    const int rowTile = blockIdx.x * (blockDim.x >> 5) + wave;
    const int mtiles  = M >> 4;                 // M is a multiple of 16 (50000 = 3125*16)
    if (rowTile >= mtiles) return;              // wave-uniform: EXEC stays all-ones

    const int   m     = lane & 15;              // M index (A) / N index (B,C)
    const int   khalf = (lane >> 4) << 1;       // 0 for lanes 0-15, 2 for lanes 16-31
    const int   row   = rowTile * 16 + m;
    const float sc    = rowScale ? rowScale[row] : 1.0f;
    const float* arow = A + (size_t)row * K;

    v8f acc0 = {}, acc1 = {}, acc2 = {}, acc3 = {};

    for (int k0 = 0; k0 < K; k0 += 4) {
        const int ka = k0 + khalf;
        v2f a;
        a.x = arow[ka]     * sc;
        a.y = arow[ka + 1] * sc;

        const float* w0 = W + (size_t)ka * EMBED_D;        // W row ka
        const float* w1 = w0 + EMBED_D;                    // W row ka+1
        v2f b0, b1, b2, b3;
        b0.x = w0[m];      b0.y = w1[m];
        b1.x = w0[16 + m]; b1.y = w1[16 + m];
        b2.x = w0[32 + m]; b2.y = w1[32 + m];
        b3.x = w0[48 + m]; b3.y = w1[48 + m];

        acc0 = __builtin_amdgcn_wmma_f32_16x16x4_f32(false, a, false, b0, (short)0, acc0, false, false);
        acc1 = __builtin_amdgcn_wmma_f32_16x16x4_f32(false, a, false, b1, (short)0, acc1, false, false);
        acc2 = __builtin_amdgcn_wmma_f32_16x16x4_f32(false, a, false, b2, (short)0, acc2, false, false);
        acc3 = __builtin_amdgcn_wmma_f32_16x16x4_f32(false, a, false, b3, (short)0, acc3, false, false);
    }

    // C/D layout: VGPR i holds row (base + i); lanes 16-31 are rows +8.
    const int col   = lane & 15;
    const int rbase = rowTile * 16 + ((lane >> 4) << 3);
#pragma unroll
    for (int i = 0; i < 8; i++) {
        float* orow = out + (size_t)(rbase + i) * EMBED_D;
        orow[col]      = acc0[i] + bias[col];
        orow[16 + col] = acc1[i] + bias[16 + col];
        orow[32 + col] = acc2[i] + bias[32 + col];
        orow[48 + col] = acc3[i] + bias[48 + col];
    }
}

// ---------------------------------------------------------------------------
// Decoder: out[i] = dot(h[i,:64], W_dec[:,0]) + b_dec. One thread per node,
// float4 (global_load_b128) reads.
// ---------------------------------------------------------------------------
__global__ void k_decode(const float* __restrict__ h, const float* __restrict__ Wd,
                         const float* __restrict__ bd, float* __restrict__ out, int M) {
    int i = blockIdx.x * blockDim.x + threadIdx.x;
    if (i < M) {
        const float4* hr = (const float4*)(h + (size_t)i * EMBED_D);
        const float4* wr = (const float4*)Wd;
        float s = 0.0f;
#pragma unroll
        for (int k = 0; k < EMBED_D / 4; k++) {
            float4 a = hr[k];
            float4 b = wr[k];
            s += a.x * b.x + a.y * b.y + a.z * b.z + a.w * b.w;
        }
        out[i] = s + bd[0];
    }
}

// ---------------------------------------------------------------------------
// Launch
// ---------------------------------------------------------------------------
extern "C" void kernel_launch(void* const* d_in, const int* in_sizes, int n_in,
                              void* d_out, int out_size, void* d_ws, size_t ws_size,
                              hipStream_t stream) {
    const float* x       = (const float*)d_in[0];  // [N,32]
    const int*   src     = (const int*)  d_in[1];  // [E]
    const int*   dst     = (const int*)  d_in[2];  // [E]
    const float* W_embed = (const float*)d_in[3];  // [32,64]
    const float* b_embed = (const float*)d_in[4];  // [64]
    const float* Ws      = (const float*)d_in[5];  // [L,64,64]
    const float* bs      = (const float*)d_in[6];  // [L,64]
    const float* W_dec   = (const float*)d_in[7];  // [64,1]
    const float* b_dec   = (const float*)d_in[8];  // [1]
    float*       out     = (float*)d_out;          // [N,1]

    const int IN_DIM = 32;
    const int N = in_sizes[0] / IN_DIM;                    // 50000 (multiple of 16)
    const int E = in_sizes[1];                             // 800000
    const int L = in_sizes[5] / (EMBED_D * EMBED_D);       // 8

    // Workspace layout (ws poisoned by harness -> zero what we need each call)
    float* h       = (float*)d_ws;                 // [N,64]  12.8 MB
    float* agg     = h + (size_t)N * EMBED_D;      // [N,64]  12.8 MB
    float* inv_out = agg + (size_t)N * EMBED_D;    // [N]
    float* inv_in  = inv_out + N;                  // [N]

    // 1) Degrees -> inverse sqrt (clamped to 1)
    hipMemsetAsync(inv_out, 0, (size_t)2 * N * sizeof(float), stream);
    k_degree<<<(E + 255) / 256, 256, 0, stream>>>(src, dst, inv_out, inv_in, E);
    k_inv_sqrt<<<(N + 255) / 256, 256, 0, stream>>>(inv_out, inv_in, N);

    // 2) Embed GEMM: h = x @ W_embed + b_embed  (K=32, no row scaling)
    const int waves_per_blk = 256 / 32;
    const int mtiles        = N / 16;
    const int gemm_blocks   = (mtiles + waves_per_blk - 1) / waves_per_blk;
    k_gemm_wmma<<<gemm_blocks, 256, 0, stream>>>(x, nullptr, W_embed, b_embed, h, N, IN_DIM);

    // 3) Message-passing layers
    const int scatter_blocks = (E * EMBED_D + 255) / 256;
    for (int l = 0; l < L; l++) {
        hipMemsetAsync(agg, 0, (size_t)N * EMBED_D * sizeof(float), stream);
        k_scatter<<<scatter_blocks, 256, 0, stream>>>(h, inv_out, src, dst, agg, E);
        k_gemm_wmma<<<gemm_blocks, 256, 0, stream>>>(agg, inv_in,
                                                     Ws + (size_t)l * EMBED_D * EMBED_D,
                                                     bs + (size_t)l * EMBED_D,
                                                     h, N, EMBED_D);
    }

    // 4) Decode
    k_decode<<<(N + 255) / 256, 256, 0, stream>>>(h, W_dec, b_dec, out, N);
}